// Qwen3Attention_62431644614816
// MI455X (gfx1250) — compile-verified
//
#include <hip/hip_runtime.h>
#include <math.h>

typedef __attribute__((ext_vector_type(2))) float v2f;
typedef __attribute__((ext_vector_type(8))) float v8f;

#define B_      32
#define KV_     4096
#define H_      4096
#define NH_     32
#define NKV_    8
#define D_      128
#define G_      4
#define QKVN    6144          // (NH + 2*NKV) * D
#define NSPLIT  16
#define CHUNK   256           // KV / NSPLIT
#define SCALE_  0.08838834764831845f   // 128^-0.5
#define EPS_    1e-6f

__device__ __forceinline__ v8f wmma4(v2f a, v2f b, v8f c) {
  // D(16x16,f32) = A(16x4,f32) x B(4x16,f32) + C
  return __builtin_amdgcn_wmma_f32_16x16x4_f32(false, a, false, b,
                                               (short)0, c, false, false);
}

// ---------------------------------------------------------------------------
// GEMM: out[32, Ntot] = A[32,4096] @ W[Ntot,4096]^T   (f32 WMMA 16x16x4)
// grid = Ntot/16 blocks, 128 threads (4 waves, K split 4x1024), LDS reduce.
// W stream prefetched 4KB ahead with high-temporal hint (weights fit in the
// 192MB L2 and persist across graph replays).
// ---------------------------------------------------------------------------
__global__ void gemm32_wmma(const float* __restrict__ A,
                            const float* __restrict__ W,
                            float* __restrict__ out,
                            int Ntot) {
  const int nt   = blockIdx.x;
  const int tid  = threadIdx.x;
  const int wave = tid >> 5;
  const int lane = tid & 31;
  const int r    = lane & 15;
  const int off  = (lane >> 4) * 2;
  const int kbase = wave * 1024;

  const float* a0 = A + (size_t)r * 4096;
  const float* a1 = A + (size_t)(r + 16) * 4096;
  const float* wr = W + (size_t)(nt * 16 + r) * 4096;

  v8f c0 = {0,0,0,0,0,0,0,0};
  v8f c1 = {0,0,0,0,0,0,0,0};
  for (int kb2 = 0; kb2 < 256; kb2 += 8) {
    __builtin_prefetch(wr + kbase + kb2 * 4 + 1024, 0, 3);
    #pragma unroll
    for (int u = 0; u < 8; ++u) {
      int k = kbase + (kb2 + u) * 4 + off;
      v2f av0 = *(const v2f*)(a0 + k);
      v2f av1 = *(const v2f*)(a1 + k);
      v2f bv  = *(const v2f*)(wr + k);
      c0 = wmma4(av0, bv, c0);
      c1 = wmma4(av1, bv, c1);
    }
  }

  __shared__ float red[4][2][8][32];
  #pragma unroll
  for (int v = 0; v < 8; ++v) {
    red[wave][0][v][lane] = c0[v];
    red[wave][1][v][lane] = c1[v];
  }
  __syncthreads();
  for (int e = tid; e < 512; e += 128) {
    int tile = e >> 8;
    int v    = (e >> 5) & 7;
    int l2   = e & 31;
    float s = red[0][tile][v][l2] + red[1][tile][v][l2] +
              red[2][tile][v][l2] + red[3][tile][v][l2];
    int m = tile * 16 + v + ((l2 < 16) ? 0 : 8);
    int n = nt * 16 + (l2 & 15);
    out[(size_t)m * Ntot + n] = s;
  }
}

// ---------------------------------------------------------------------------
// Per-head RMSNorm + NeoX RoPE for q (32 heads) and k (8 heads) rows.
// One wave per row; 4 rows per 128-thread block; 1280 rows total.
// ---------------------------------------------------------------------------
__global__ void norm_rope(const float* __restrict__ qkv,
                          const float* __restrict__ qw,
                          const float* __restrict__ kw,
                          const int* __restrict__ pos_ids,
                          float* __restrict__ q_ws,
                          float* __restrict__ k_ws) {
  const int wave = threadIdx.x >> 5;
  const int lane = threadIdx.x & 31;
  const int row  = blockIdx.x * 4 + wave;       // 0..1279
  const int b    = row / 40;
  const int h    = row % 40;                    // <32: q head, else k head
  const float* src = qkv + (size_t)b * QKVN + h * 128;
  const float* wv  = (h < 32) ? qw : kw;

  float x[4], w4[4];
  float ss = 0.f;
  #pragma unroll
  for (int j = 0; j < 4; ++j) {
    x[j]  = src[lane * 4 + j];
    w4[j] = wv[lane * 4 + j];
    ss += x[j] * x[j];
  }
  #pragma unroll
  for (int m = 16; m >= 1; m >>= 1) ss += __shfl_xor(ss, m);
  float rs  = rsqrtf(ss * (1.0f / 128.0f) + EPS_);
  float pos = (float)pos_ids[b];
  float* dst = (h < 32) ? (q_ws + (size_t)b * 4096 + h * 128)
                        : (k_ws + (size_t)b * 1024 + (h - 32) * 128);
  const float LNTH = 13.815510557964274f / 64.0f;   // ln(1e6)/64
  #pragma unroll
  for (int j = 0; j < 4; ++j) {
    float xn    = x[j] * rs * w4[j];
    float other = __shfl_xor(xn, 16);              // x2 (or x1) partner
    int   jidx  = (lane & 15) * 4 + j;             // rotary index 0..63
    float ang   = pos * __expf(-LNTH * (float)jidx);
    float s, c;
    __sincosf(ang, &s, &c);
    float o = (lane < 16) ? (xn * c - other * s)   // first half
                          : (xn * c + other * s);  // second half
    dst[lane * 4 + j] = o;
  }
}

// ---------------------------------------------------------------------------
// Flash-decode GQA attention, transposed WMMA formulation.
// grid = B*NKV*NSPLIT blocks, 32 threads (1 wave). Chunk of 256 positions.
//   scores^T[s,g] = sum_d K[s,d] * q[g,d]   (A=K rows, B=q^T)
//   o^T[d,g]     += sum_s V[s,d] * p^T[s,g] (A=V^T,   B=p^T)
// qrow g == lane%16 everywhere -> per-lane online softmax state.
// New-token k/v substituted where s == out_cache_loc[b] (inputs untouched).
// K/V streams prefetched non-temporally (1GB of KV can't live in L2).
// ---------------------------------------------------------------------------
__global__ void attn_decode(const float* __restrict__ q_ws,
                            const float* __restrict__ k_ws,
                            const float* __restrict__ qkv,
                            const float* __restrict__ kbuf,
                            const float* __restrict__ vbuf,
                            const int* __restrict__ cache_loc,
                            float* __restrict__ o_part,
                            float* __restrict__ m_part,
                            float* __restrict__ l_part) {
  const int split = blockIdx.x & (NSPLIT - 1);
  const int h     = (blockIdx.x / NSPLIT) & (NKV_ - 1);
  const int b     = blockIdx.x / (NSPLIT * NKV_);
  const int lane  = threadIdx.x;
  const int half  = lane >> 4;
  const int r     = lane & 15;
  const int off   = half * 2;
  const int loc   = cache_loc[b];
  const int s0    = split * CHUNK;

  // q^T B-fragments resident in registers (rows 0..3 valid, rest zero-padded)
  const v2f vzero = {0.f, 0.f};
  v2f qb[32];
  const float* qrow = q_ws + (size_t)b * 4096 + (h * 4 + r) * 128;
  #pragma unroll
  for (int ks = 0; ks < 32; ++ks) {
    v2f qv = vzero;
    if (r < 4) qv = *(const v2f*)(qrow + ks * 4 + off);
    qb[ks] = qv;
  }

  const float* kb_base = kbuf + (size_t)b * KV_ * NKV_ * 128 + h * 128;
  const float* vb_base = vbuf + (size_t)b * KV_ * NKV_ * 128 + h * 128;
  const float* knew = k_ws + (size_t)b * 1024 + h * 128;
  const float* vnew = qkv + (size_t)b * QKVN + (NH_ + NKV_) * 128 + h * 128;

  v8f oc[8];
  #pragma unroll
  for (int i = 0; i < 8; ++i) oc[i] = (v8f){0,0,0,0,0,0,0,0};
  float m_run = -1e30f, l_run = 0.f;

  for (int t = 0; t < 16; ++t) {
    int sbase = s0 + t * 16;

    // Prefetch: current V tile (16 rows x 512B = 64 lines of 128B) and the
    // next K tile, 2 lines per lane, non-temporal.
    {
      int row0 = (lane >> 2), lo0 = (lane & 3) * 32;
      int row1 = ((lane + 32) >> 2), lo1 = ((lane + 32) & 3) * 32;
      __builtin_prefetch(vb_base + (size_t)(sbase + row0) * 1024 + lo0, 0, 0);
      __builtin_prefetch(vb_base + (size_t)(sbase + row1) * 1024 + lo1, 0, 0);
      if (t < 15) {
        __builtin_prefetch(kb_base + (size_t)(sbase + 16 + row0) * 1024 + lo0, 0, 0);
        __builtin_prefetch(kb_base + (size_t)(sbase + 16 + row1) * 1024 + lo1, 0, 0);
      }
    }

    // ---- scores^T tile: 16 positions x 16 qrows ----
    int sA = sbase + r;
    const float* kp = (sA == loc) ? knew
                                  : (kb_base + (size_t)sA * (NKV_ * 128));
    v8f ca = (v8f){0,0,0,0,0,0,0,0};
    v8f cb = (v8f){0,0,0,0,0,0,0,0};
    #pragma unroll
    for (int ks = 0; ks < 32; ks += 2) {
      v2f a0 = *(const v2f*)(kp + ks * 4 + off);
      v2f a1 = *(const v2f*)(kp + (ks + 1) * 4 + off);
      ca = wmma4(a0, qb[ks], ca);
      cb = wmma4(a1, qb[ks + 1], cb);
    }
    float p[8];
    float tm = -1e30f;
    #pragma unroll
    for (int v = 0; v < 8; ++v) {
      p[v] = (ca[v] + cb[v]) * SCALE_;
      tm = fmaxf(tm, p[v]);
    }
    tm = fmaxf(tm, __shfl_xor(tm, 16));
    float m_new = fmaxf(m_run, tm);
    float sf = __expf(m_run - m_new);
    #pragma unroll
    for (int i = 0; i < 8; ++i)
      #pragma unroll
      for (int v = 0; v < 8; ++v) oc[i][v] *= sf;
    float psum = 0.f;
    #pragma unroll
    for (int v = 0; v < 8; ++v) { p[v] = __expf(p[v] - m_new); psum += p[v]; }
    psum += __shfl_xor(psum, 16);
    l_run = l_run * sf + psum;
    m_run = m_new;

    // ---- PV: o^T += V^T x p^T ----
    #pragma unroll
    for (int kk = 0; kk < 4; ++kk) {
      v2f bfr;
      if (kk == 0) {
        bfr.x = half ? __shfl_xor(p[2], 16) : p[0];
        bfr.y = half ? __shfl_xor(p[3], 16) : p[1];
      } else if (kk == 1) {
        bfr.x = half ? __shfl_xor(p[6], 16) : p[4];
        bfr.y = half ? __shfl_xor(p[7], 16) : p[5];
      } else if (kk == 2) {
        bfr.x = half ? p[2] : __shfl_xor(p[0], 16);
        bfr.y = half ? p[3] : __shfl_xor(p[1], 16);
      } else {
        bfr.x = half ? p[6] : __shfl_xor(p[4], 16);
        bfr.y = half ? p[7] : __shfl_xor(p[5], 16);
      }
      int sx = sbase + kk * 4 + off;
      int sy = sx + 1;
      const float* vpx = (sx == loc) ? vnew
                                     : (vb_base + (size_t)sx * (NKV_ * 128));
      const float* vpy = (sy == loc) ? vnew
                                     : (vb_base + (size_t)sy * (NKV_ * 128));
      #pragma unroll
      for (int dt = 0; dt < 8; ++dt) {
        v2f a;
        a.x = vpx[dt * 16 + r];
        a.y = vpy[dt * 16 + r];
        oc[dt] = wmma4(a, bfr, oc[dt]);
      }
    }
  }

  // write partials: o^T element (dt,v,lane): d = dt*16 + v + half*8, g = r
  size_t pbase = ((size_t)(b * NKV_ + h) * NSPLIT + split) * G_;
  if (r < 4) {
    #pragma unroll
    for (int dt = 0; dt < 8; ++dt)
      #pragma unroll
      for (int v = 0; v < 8; ++v) {
        int d = dt * 16 + v + half * 8;
        o_part[(pbase + r) * 128 + d] = oc[dt][v];
      }
    if (half == 0) {
      m_part[pbase + r] = m_run;
      l_part[pbase + r] = l_run;
    }
  }
}

// ---------------------------------------------------------------------------
// Merge NSPLIT flash partials. grid = B*NH, block = 128 (d index).
// ---------------------------------------------------------------------------
__global__ void combine(const float* __restrict__ o_part,
                        const float* __restrict__ m_part,
                        const float* __restrict__ l_part,
                        float* __restrict__ o_fin) {
  int g = blockIdx.x & 3;
  int h = (blockIdx.x >> 2) & 7;
  int b = blockIdx.x >> 5;
  int d = threadIdx.x;
  size_t base = ((size_t)(b * NKV_ + h) * NSPLIT) * G_ + g;
  float M = -1e30f;
  for (int i = 0; i < NSPLIT; ++i)
    M = fmaxf(M, m_part[base + (size_t)i * G_]);
  float L = 0.f, acc = 0.f;
  for (int i = 0; i < NSPLIT; ++i) {
    float w = __expf(m_part[base + (size_t)i * G_] - M);
    L   += l_part[base + (size_t)i * G_] * w;
    acc += w * o_part[(base + (size_t)i * G_) * 128 + d];
  }
  o_fin[(size_t)b * 4096 + (h * 4 + g) * 128 + d] = acc / L;
}

// ---------------------------------------------------------------------------
extern "C" void kernel_launch(void* const* d_in, const int* in_sizes, int n_in,
                              void* d_out, int out_size, void* d_ws, size_t ws_size,
                              hipStream_t stream) {
  const int*   pos    = (const int*)d_in[0];
  const float* hidden = (const float*)d_in[1];
  const float* kbuf   = (const float*)d_in[2];
  const float* vbuf   = (const float*)d_in[3];
  const int*   loc    = (const int*)d_in[4];
  const float* wqkv   = (const float*)d_in[5];
  const float* wo     = (const float*)d_in[6];
  const float* qnw    = (const float*)d_in[7];
  const float* knw    = (const float*)d_in[8];
  float* out = (float*)d_out;

  float* qkv_ws = (float*)d_ws;                       // 32*6144
  float* q_ws   = qkv_ws + 32 * 6144;                 // 32*32*128
  float* k_ws   = q_ws   + 32 * 32 * 128;             // 32*8*128
  float* o_part = k_ws   + 32 * 8 * 128;              // 32*8*16*4*128
  float* m_part = o_part + 32 * 8 * 16 * 4 * 128;     // 32*8*16*4
  float* l_part = m_part + 32 * 8 * 16 * 4;           // 32*8*16*4
  float* o_fin  = l_part + 32 * 8 * 16 * 4;           // 32*4096

  gemm32_wmma<<<384, 128, 0, stream>>>(hidden, wqkv, qkv_ws, QKVN);
  norm_rope<<<320, 128, 0, stream>>>(qkv_ws, qnw, knw, pos, q_ws, k_ws);
  attn_decode<<<B_ * NKV_ * NSPLIT, 32, 0, stream>>>(q_ws, k_ws, qkv_ws,
                                                     kbuf, vbuf, loc,
                                                     o_part, m_part, l_part);
  combine<<<B_ * NH_, 128, 0, stream>>>(o_part, m_part, l_part, o_fin);
  gemm32_wmma<<<256, 128, 0, stream>>>(o_fin, wo, out, H_);
}